// SE_Block_59004260712650
// MI455X (gfx1250) — compile-verified
//
#include <hip/hip_runtime.h>
#include <hip/hip_bf16.h>
#include <math.h>

// Problem constants (from reference): B=64, T=5, V=1024, F=512
#define NGROUP   320         // B*T
#define VF       524288      // V*F (2^19)
#define VF4      131072      // VF/4 (2^17)
#define BPG      128         // reduction blocks per (b,t) group
#define ELEM_PB  4096        // elements reduced per block (= VF/BPG)
#define N4_TOTAL 41943040ULL // B*T*V*F / 4 float4s

typedef __attribute__((ext_vector_type(2))) float v2f;
typedef __attribute__((ext_vector_type(8))) float v8f;

__device__ __forceinline__ v8f wmma_f32_k4(v2f a, v2f b, v8f c) {
    // D = A(16x4,f32) * B(4x16,f32) + C(16x16,f32); wave32, EXEC must be all 1s.
    return __builtin_amdgcn_wmma_f32_16x16x4_f32(
        /*neg_a=*/false, a, /*neg_b=*/false, b,
        /*c_mod=*/(short)0, c, /*reuse_a=*/false, /*reuse_b=*/false);
}

// ---------------------------------------------------------------------------
// Pass 1: deterministic partial reduction. Each block reduces 4096 contiguous
// floats (one 1/128th slice of one (b,t) group) to one partial sum + max.
// ---------------------------------------------------------------------------
__global__ void reduce_partial(const float4* __restrict__ x4,
                               float* __restrict__ psum,
                               float* __restrict__ pmax) {
    const int tid = threadIdx.x;
    const size_t base = (size_t)blockIdx.x * 1024;  // float4 units (4096 floats)

    float s = 0.f;
    float m = -__builtin_inff();
#pragma unroll
    for (int j = 0; j < 4; ++j) {
        float4 v = x4[base + (size_t)j * 256 + tid];   // coalesced global_load_b128
        s += (v.x + v.y) + (v.z + v.w);
        m = fmaxf(m, fmaxf(fmaxf(v.x, v.y), fmaxf(v.z, v.w)));
    }
    // wave32 tree reduce (fixed order -> deterministic)
#pragma unroll
    for (int off = 16; off > 0; off >>= 1) {
        s += __shfl_xor(s, off, 32);
        m = fmaxf(m, __shfl_xor(m, off, 32));
    }
    __shared__ float ls[8], lm[8];
    const int lane = tid & 31, wv = tid >> 5;
    if (lane == 0) { ls[wv] = s; lm[wv] = m; }
    __syncthreads();
    if (tid == 0) {
        float S = ls[0], M = lm[0];
#pragma unroll
        for (int i = 1; i < 8; ++i) { S += ls[i]; M = fmaxf(M, lm[i]); }
        psum[blockIdx.x] = S;   // blockIdx.x == group*BPG + slice (contiguous)
        pmax[blockIdx.x] = M;
    }
}

// ---------------------------------------------------------------------------
// Pass 2: finalize pools + WMMA gate MLP (fp32 WMMA 16x16x4). One block,
// 256 threads = 8 wave32s. Waves 0..3: avg branch (M-tiles 0..3);
// waves 4..7: max branch. Writes se[320] = (1-w)*s_avg + w*s_max.
// ---------------------------------------------------------------------------
__global__ void gate_wmma(const float* __restrict__ psum,
                          const float* __restrict__ pmax,
                          const float* __restrict__ gW1,  // [5][25]
                          const float* __restrict__ gb1,  // [25]
                          const float* __restrict__ gW2,  // [25][5]
                          const float* __restrict__ gb2,  // [5]
                          const float* __restrict__ gw,   // scalar
                          float* __restrict__ g_se) {     // [320]
    __shared__ float lds_P[2][64 * 8];    // pooled inputs, K padded 5->8
    __shared__ float lds_H[2][64 * 32];   // hidden, K padded 25->32 (zeros)
    __shared__ float lds_S[2][64 * 5];    // sigmoid outputs

    const int tid = threadIdx.x;

    // zero-pad pooled tiles
    for (int i = tid; i < 2 * 64 * 8; i += 256) (&lds_P[0][0])[i] = 0.f;
    __syncthreads();

    // fold the 128 partials per group, fixed order -> deterministic
    for (int g = tid; g < NGROUP; g += 256) {
        float S = 0.f, M = -__builtin_inff();
        const int base = g * BPG;
        for (int p = 0; p < BPG; ++p) {
            S += psum[base + p];
            M = fmaxf(M, pmax[base + p]);
        }
        const int row = g / 5, col = g % 5;            // row = batch, col = T channel
        lds_P[0][row * 8 + col] = S * (1.0f / (float)VF);
        lds_P[1][row * 8 + col] = M;
    }
    __syncthreads();

    const int wave   = tid >> 5;   // 0..7
    const int lane   = tid & 31;
    const int branch = wave >> 2;  // 0 = avg, 1 = max
    const int mtile  = wave & 3;   // 16-row tile of the 64-batch dim
    const int Mlo    = lane & 15;  // A: M%16 ; B/C: N
    const int hi     = lane >> 4;  // half-wave select

    // ---- Layer 1: H = relu(P[64,5] @ W1[5,25] + b1), padded K=8, N=32 ----
    for (int nt = 0; nt < 2; ++nt) {
        v8f acc = {};
#pragma unroll
        for (int s = 0; s < 2; ++s) {
            const int ka = s * 4 + hi * 2;             // A layout: K = v + 2*hi
            v2f a, b;
            a.x = lds_P[branch][(mtile * 16 + Mlo) * 8 + ka];
            a.y = lds_P[branch][(mtile * 16 + Mlo) * 8 + ka + 1];
            const int n = nt * 16 + Mlo;
            b.x = (ka     < 5 && n < 25) ? gW1[ka * 25 + n]       : 0.f;
            b.y = (ka + 1 < 5 && n < 25) ? gW1[(ka + 1) * 25 + n] : 0.f;
            acc = wmma_f32_k4(a, b, acc);
        }
        const int n = nt * 16 + Mlo;
        const float bias = (n < 25) ? gb1[n] : 0.f;
#pragma unroll
        for (int v = 0; v < 8; ++v) {                  // C layout: M = v + 8*hi
            const int row = mtile * 16 + hi * 8 + v;
            float h = acc[v] + bias;
            lds_H[branch][row * 32 + n] = (n < 25) ? fmaxf(h, 0.f) : 0.f;
        }
    }
    __syncthreads();

    // ---- Layer 2: S = sigmoid(H[64,25] @ W2[25,5] + b2), K padded 25->28 ----
    v8f acc2 = {};
#pragma unroll
    for (int s = 0; s < 7; ++s) {
        const int ka = s * 4 + hi * 2;                 // max 27 < 32 (padded with 0)
        v2f a, b;
        const int row = mtile * 16 + Mlo;
        a.x = lds_H[branch][row * 32 + ka];
        a.y = lds_H[branch][row * 32 + ka + 1];
        b.x = (ka     < 25 && Mlo < 5) ? gW2[ka * 5 + Mlo]       : 0.f;
        b.y = (ka + 1 < 25 && Mlo < 5) ? gW2[(ka + 1) * 5 + Mlo] : 0.f;
        acc2 = wmma_f32_k4(a, b, acc2);
    }
    if (Mlo < 5) {
        const float bias = gb2[Mlo];
#pragma unroll
        for (int v = 0; v < 8; ++v) {
            const int row = mtile * 16 + hi * 8 + v;
            const float z = acc2[v] + bias;
            lds_S[branch][row * 5 + Mlo] = 1.0f / (1.0f + __expf(-z));
        }
    }
    __syncthreads();

    if (tid < NGROUP) {
        const float wv = gw[0];
        g_se[tid] = (1.0f - wv) * lds_S[0][tid] + wv * lds_S[1][tid];
    }
}

// ---------------------------------------------------------------------------
// Pass 3: out = x * se[bt], pure streaming, float4 (b128) loads/stores.
// ---------------------------------------------------------------------------
__global__ void scale4(const float4* __restrict__ x4,
                       const float* __restrict__ se,
                       float4* __restrict__ out4,
                       size_t n4) {
    const size_t stride = (size_t)gridDim.x * blockDim.x;
    size_t i = (size_t)blockIdx.x * blockDim.x + threadIdx.x;
    for (; i < n4; i += stride) {
        if (i + stride < n4) __builtin_prefetch(&x4[i + stride], 0, 0);
        const float s = se[i >> 17];                   // bt = (4*i) / (V*F)
        float4 v = x4[i];
        v.x *= s; v.y *= s; v.z *= s; v.w *= s;
        out4[i] = v;
    }
}

extern "C" void kernel_launch(void* const* d_in, const int* in_sizes, int n_in,
                              void* d_out, int out_size, void* d_ws, size_t ws_size,
                              hipStream_t stream) {
    const float* x  = (const float*)d_in[0];
    const float* W1 = (const float*)d_in[1];
    const float* b1 = (const float*)d_in[2];
    const float* W2 = (const float*)d_in[3];
    const float* b2 = (const float*)d_in[4];
    const float* w  = (const float*)d_in[5];
    float* out = (float*)d_out;

    float* ws   = (float*)d_ws;
    float* psum = ws;                       // [320*128]
    float* pmax = ws + NGROUP * BPG;        // [320*128]
    float* se   = ws + 2 * NGROUP * BPG;    // [320]

    reduce_partial<<<NGROUP * BPG, 256, 0, stream>>>((const float4*)x, psum, pmax);
    gate_wmma<<<1, 256, 0, stream>>>(psum, pmax, W1, b1, W2, b2, w, se);
    scale4<<<8192, 256, 0, stream>>>((const float4*)x, se, (float4*)out, (size_t)N4_TOTAL);
}